// SingleHeadAttention_19387482374359
// MI455X (gfx1250) — compile-verified
//
#include <hip/hip_runtime.h>
#include <math.h>

// Problem constants (from reference): B=4, S=2048, E=512, H=8, d=64
#define BB 4
#define SS 2048
#define EE 512
#define HH 8
#define DD 64
#define KTOP 682            // S // 3
#define ABS_THRESH 0.6f
#define PSTR 2052           // padded LDS row stride for score strip (2052 % 64 == 4 -> conflict-free)

typedef float v2f __attribute__((ext_vector_type(2)));
typedef float v8f __attribute__((ext_vector_type(8)));
typedef int   v4i __attribute__((ext_vector_type(4)));

#if defined(__HIP_DEVICE_COMPILE__) && __has_builtin(__builtin_amdgcn_global_load_async_to_lds_b128)
#define HAVE_ASYNC_LDS 1
typedef __attribute__((address_space(1))) v4i* gv4i_p;   // global int4*
typedef __attribute__((address_space(3))) v4i* lv4i_p;   // LDS int4*
#else
#define HAVE_ASYNC_LDS 0
#endif

// ---------------------------------------------------------------------------
// Stage one 16x64 f32 tile (row stride EE in global) into an LDS tile with
// row stride 68 floats. One wave does the whole tile.
// Async path: GLOBAL_LOAD_ASYNC_TO_LDS_B128 (ASYNCcnt), else float4 via VGPRs.
// ---------------------------------------------------------------------------
__device__ __forceinline__ void stage_tile(const float* gbase, float* lbase, int lane) {
  const int rh = lane >> 4;            // 0/1: two rows per pass
  const int c4 = (lane & 15) * 4;      // 16 lanes cover 64 columns
#pragma unroll
  for (int r2 = 0; r2 < 16; r2 += 2) {
    const int r = r2 + rh;
    const float* g = gbase + (size_t)r * EE + c4;
    float* l = lbase + r * 68 + c4;
#if HAVE_ASYNC_LDS
    __builtin_amdgcn_global_load_async_to_lds_b128((gv4i_p)g, (lv4i_p)l, 0, 0);
#else
    *(float4*)l = *(const float4*)g;
#endif
  }
}

__device__ __forceinline__ void stage_wait() {
#if HAVE_ASYNC_LDS
#if __has_builtin(__builtin_amdgcn_s_wait_asynccnt)
  __builtin_amdgcn_s_wait_asynccnt(0);
#else
  asm volatile("s_wait_asynccnt 0x0" ::: "memory");
#endif
#endif
}

// ---------------------------------------------------------------------------
// Kernel 1: per-(b,h,s) RECIPROCAL L2 norms of Q rows and K rows -> workspace
// (reciprocals precomputed here so the hot WMMA loop is pure v_mul)
// ---------------------------------------------------------------------------
__global__ __launch_bounds__(256) void topk_attn_norms_kernel(
    const float* __restrict__ Qg, const float* __restrict__ Kg,
    float* __restrict__ rnq, float* __restrict__ rnk) {
  int gtid = blockIdx.x * 256 + threadIdx.x;
  int wid  = gtid >> 5;          // global wave id, [0, 2*B*H*S)
  int lane = gtid & 31;
  int which = wid >> 16;         // 0 -> Q, 1 -> K   (B*H*S = 65536 rows each)
  int row   = wid & 65535;       // row = (b*H + h)*S + s
  int s  = row & (SS - 1);
  int bh = row >> 11;
  int h  = bh & (HH - 1);
  int b  = bh >> 3;
  const float* src = which ? Kg : Qg;
  const float* p = src + ((size_t)(b * SS + s)) * EE + (size_t)h * DD;
  float x0 = p[lane], x1 = p[lane + 32];
  float sum = x0 * x0 + x1 * x1;
#pragma unroll
  for (int off = 16; off > 0; off >>= 1) sum += __shfl_xor(sum, off, 32);
  if (lane == 0) (which ? rnk : rnq)[row] = 1.0f / sqrtf(sum);
}

// ---------------------------------------------------------------------------
// Kernel 2: fused top-k attention for one (b, h, 16-row q tile)
// 256 threads = 8 waves. Score strip 16 x 2048 lives in LDS (~128 KB of the
// 320 KB WGP LDS). Histogram select instead of bisection: 2 passes not ~20.
// ---------------------------------------------------------------------------
__global__ __launch_bounds__(256) void topk_attn_main_kernel(
    const float* __restrict__ Qg, const float* __restrict__ Kg,
    const float* __restrict__ Vg, const float* __restrict__ rnq,
    const float* __restrict__ rnk, float* __restrict__ Og) {
  __shared__ float sP[16 * PSTR];       // score strip / masked P   (~128 KB)
  __shared__ float sQ[16][68];          // Q tile (padded)
  __shared__ float sK[8][16][68];       // per-wave K/V staging tiles
  __shared__ float sAcc[8][16][68];     // per-wave PV partial sums
  __shared__ int   sHist[16][256];      // per-row histogram
  __shared__ float sRed[16][16];        // deterministic L1 reduction scratch
  __shared__ float sRq[16];             // 1 / |q_row|
  __shared__ float sLo[16];
  __shared__ float sBias[16];
  __shared__ float sInvL1[16];
  __shared__ int   sBkt[16];
  __shared__ int   sNeed[16];

  const int tid  = threadIdx.x;
  const int lane = tid & 31;
  const int w    = tid >> 5;            // wave id 0..7
  const int wg   = blockIdx.x;
  const int qt   = wg & 127;            // S/16 = 128 q tiles
  const int h    = (wg >> 7) & (HH - 1);
  const int b    = wg >> 10;
  const int q0   = qt * 16;

  const size_t headoff = (size_t)h * DD;
  const float* Qh   = Qg + ((size_t)b * SS) * EE + headoff;
  const float* Kh   = Kg + ((size_t)b * SS) * EE + headoff;
  const float* Vh   = Vg + ((size_t)b * SS) * EE + headoff;
  const float* rnqh = rnq + ((size_t)(b * HH + h)) * SS;
  const float* rnkh = rnk + ((size_t)(b * HH + h)) * SS;

  // ---- load Q tile (16x64 f32, coalesced float4) + 1/|q| -------------------
  {
    int r  = tid >> 4;
    int c4 = (tid & 15) * 4;
    float4 val = *(const float4*)(Qh + (size_t)(q0 + r) * EE + c4);
    *(float4*)&sQ[r][c4] = val;
  }
  if (tid < 16) sRq[tid] = rnqh[q0 + tid];
  __syncthreads();

  const int m     = lane & 15;          // M index for A frag / N index for B frag
  const int koff  = (lane >> 4) * 2;    // K sub-offset per half-wave (16x16x4 layout)
  const int rbase = (lane >> 4) * 8;    // C/D row base per half-wave

  // ---- Phase 1: scores = (Q K^T)/(|q||k|), wave w owns K chunks t*8+w ------
  for (int t = 0; t < 16; ++t) {
    const int kb = (t * 8 + w) * 16;
    stage_tile(Kh + (size_t)kb * EE, &sK[w][0][0], lane);
    if (t < 15) {
      int kbn = ((t + 1) * 8 + w) * 16;
      __builtin_prefetch((const char*)(Kh + (size_t)(kbn + m) * EE + (lane >> 4) * 32), 0, 0);
    }
    stage_wait();
    __syncthreads();

    // two independent 8-deep WMMA chains for ILP
    v8f c0 = {}, c1 = {};
#pragma unroll
    for (int k0 = 0; k0 < 32; k0 += 4) {
      v2f a, bb;
      a.x  = sQ[m][k0 + koff];
      a.y  = sQ[m][k0 + koff + 1];
      bb.x = sK[w][m][k0 + koff];       // B[kd][n] = K[kb+n][kd], n == m
      bb.y = sK[w][m][k0 + koff + 1];
      c0 = __builtin_amdgcn_wmma_f32_16x16x4_f32(false, a, false, bb,
                                                 (short)0, c0, false, false);
      v2f a2, bb2;
      a2.x  = sQ[m][32 + k0 + koff];
      a2.y  = sQ[m][32 + k0 + koff + 1];
      bb2.x = sK[w][m][32 + k0 + koff];
      bb2.y = sK[w][m][32 + k0 + koff + 1];
      c1 = __builtin_amdgcn_wmma_f32_16x16x4_f32(false, a2, false, bb2,
                                                 (short)0, c1, false, false);
    }
    // scale by (1/|q_row|) * (1/|k_col|) and scatter to score strip
    const float rkcol = rnkh[kb + m];
#pragma unroll
    for (int j = 0; j < 8; ++j) {
      int row = rbase + j;
      sP[row * PSTR + kb + m] = (c0[j] + c1[j]) * (sRq[row] * rkcol);
    }
  }
  __syncthreads();

  // ---- Phase 2a: coarse 256-bucket histogram over [-1, 1] per row ----------
  for (int i = tid; i < 16 * 256; i += 256) ((int*)sHist)[i] = 0;
  __syncthreads();
  {
    int r = tid >> 4;
    for (int cidx = (tid & 15); cidx < SS; cidx += 16) {
      float sv = sP[r * PSTR + cidx];
      int b1 = (int)floorf((sv + 1.0f) * 128.0f);
      b1 = min(255, max(0, b1));
      atomicAdd(&sHist[r][b1], 1);
    }
  }
  __syncthreads();
  if (tid < 16) {
    int cum = 0, bidx = 0, need = KTOP;
    for (int j = 255; j >= 0; --j) {
      int cj = sHist[tid][j];
      if (cum + cj >= KTOP) { bidx = j; need = KTOP - cum; break; }
      cum += cj;
    }
    sBkt[tid]  = bidx;
    sNeed[tid] = need;
    sLo[tid]   = (float)bidx * (1.0f / 128.0f) - 1.0f;
  }
  __syncthreads();

  // ---- Phase 2b: refine inside the boundary bucket (res ~ 3e-5) ------------
  for (int i = tid; i < 16 * 256; i += 256) ((int*)sHist)[i] = 0;
  __syncthreads();
  {
    int r = tid >> 4;
    int bidx = sBkt[r];
    float lo = sLo[r];
    for (int cidx = (tid & 15); cidx < SS; cidx += 16) {
      float sv = sP[r * PSTR + cidx];
      int b1 = (int)floorf((sv + 1.0f) * 128.0f);
      b1 = min(255, max(0, b1));
      if (b1 == bidx) {
        int b2 = (int)floorf((sv - lo) * 32768.0f);
        b2 = min(255, max(0, b2));
        atomicAdd(&sHist[r][b2], 1);
      }
    }
  }
  __syncthreads();
  if (tid < 16) {
    int need = sNeed[tid], cum = 0, b2 = 0;
    for (int j = 255; j >= 0; --j) {
      cum += sHist[tid][j];
      if (cum >= need) { b2 = j; break; }
    }
    sBias[tid] = sLo[tid] + (float)b2 * (1.0f / 32768.0f);
  }
  __syncthreads();

  // ---- Phase 2c: mask scores, deterministic L1 row sums --------------------
  {
    int r = tid >> 4;
    float bias = sBias[r];
    float acc = 0.0f;
    for (int cidx = (tid & 15); cidx < SS; cidx += 16) {
      float sv = sP[r * PSTR + cidx];
      bool keep = (sv >= bias) && (fabsf(sv) >= ABS_THRESH);
      float p = keep ? sv : 0.0f;
      sP[r * PSTR + cidx] = p;
      acc += fabsf(p);
    }
    sRed[r][tid & 15] = acc;
  }
  __syncthreads();
  if (tid < 16) {
    float ssum = 0.0f;
#pragma unroll
    for (int j = 0; j < 16; ++j) ssum += sRed[tid][j];
    sInvL1[tid] = 1.0f / ssum;
  }
  __syncthreads();

  // ---- Phase 3: out = P V, wave-private accumulators, 4 WMMA chains --------
  v8f acc0 = {}, acc1 = {}, acc2 = {}, acc3 = {};
  for (int t = 0; t < 16; ++t) {
    const int kb = (t * 8 + w) * 16;
    stage_tile(Vh + (size_t)kb * EE, &sK[w][0][0], lane);
    if (t < 15) {
      int kbn = ((t + 1) * 8 + w) * 16;
      __builtin_prefetch((const char*)(Vh + (size_t)(kbn + m) * EE + (lane >> 4) * 32), 0, 0);
    }
    stage_wait();
    __syncthreads();

#pragma unroll
    for (int k0 = 0; k0 < 16; k0 += 4) {
      v2f a;
      a.x = sP[m * PSTR + kb + k0 + koff];
      a.y = sP[m * PSTR + kb + k0 + koff + 1];
      int kr0 = k0 + koff, kr1 = k0 + koff + 1;
      v2f b0, b1, b2, b3;
      b0.x = sK[w][kr0][ 0 + m]; b0.y = sK[w][kr1][ 0 + m];
      b1.x = sK[w][kr0][16 + m]; b1.y = sK[w][kr1][16 + m];
      b2.x = sK[w][kr0][32 + m]; b2.y = sK[w][kr1][32 + m];
      b3.x = sK[w][kr0][48 + m]; b3.y = sK[w][kr1][48 + m];
      acc0 = __builtin_amdgcn_wmma_f32_16x16x4_f32(false, a, false, b0, (short)0, acc0, false, false);
      acc1 = __builtin_amdgcn_wmma_f32_16x16x4_f32(false, a, false, b1, (short)0, acc1, false, false);
      acc2 = __builtin_amdgcn_wmma_f32_16x16x4_f32(false, a, false, b2, (short)0, acc2, false, false);
      acc3 = __builtin_amdgcn_wmma_f32_16x16x4_f32(false, a, false, b3, (short)0, acc3, false, false);
    }
  }
  // dump wave partials
#pragma unroll
  for (int j = 0; j < 8; ++j) {
    int row = rbase + j;
    sAcc[w][row][ 0 + m] = acc0[j];
    sAcc[w][row][16 + m] = acc1[j];
    sAcc[w][row][32 + m] = acc2[j];
    sAcc[w][row][48 + m] = acc3[j];
  }
  __syncthreads();

  // ---- Phase 4: deterministic cross-wave reduce, L1 normalize, store -------
  float* Oh = Og + ((size_t)b * SS) * EE + headoff;
  for (int i = tid; i < 16 * DD; i += 256) {
    int r  = i >> 6;
    int cc = i & 63;
    float ssum = 0.0f;
#pragma unroll
    for (int ww = 0; ww < 8; ++ww) ssum += sAcc[ww][r][cc];
    Oh[(size_t)(q0 + r) * EE + cc] = ssum * sInvL1[r];
  }
}

// ---------------------------------------------------------------------------
extern "C" void kernel_launch(void* const* d_in, const int* in_sizes, int n_in,
                              void* d_out, int out_size, void* d_ws, size_t ws_size,
                              hipStream_t stream) {
  const float* Qg = (const float*)d_in[0];
  const float* Kg = (const float*)d_in[1];
  const float* Vg = (const float*)d_in[2];
  float* Og = (float*)d_out;

  float* rnq = (float*)d_ws;                 // B*H*S = 65536 floats (1/|q|)
  float* rnk = rnq + (BB * HH * SS);         // another 65536 floats (1/|k|)

  // 2 * B*H*S rows, one wave32 per row, 8 waves per block
  int norm_blocks = (2 * BB * HH * SS) / 8;
  topk_attn_norms_kernel<<<norm_blocks, 256, 0, stream>>>(Qg, Kg, rnq, rnk);

  // one block per (b, h, 16-row q tile): 4 * 8 * 128 = 4096 blocks
  topk_attn_main_kernel<<<BB * HH * (SS / 16), 256, 0, stream>>>(
      Qg, Kg, Vg, rnq, rnk, Og);
}